// HeteroGraphNN_56693568307502
// MI455X (gfx1250) — compile-verified
//
#include <hip/hip_runtime.h>
#include <cstdint>
#include <cstddef>

// ---------------- CDNA5 types ----------------
typedef float v2f __attribute__((ext_vector_type(2)));
typedef float v8f __attribute__((ext_vector_type(8)));

// Native fp32 global atomic add (global_atomic_add_f32 on gfx1250)
static __device__ __forceinline__ void atomAddF(float* p, float v) {
  unsafeAtomicAdd(p, v);
}

// ---------------- static graph metadata ----------------
#define NEDGE 150000
#define NTYPES 5
#define TOTAL_NODES 116000
#define HID 128
#define TOTAL_LEN (TOTAL_NODES * HID)   // 14,848,000

// ---------------- kernels ----------------

// h0[v][c] = sum_k feat[v][k] * W[k][c] + b[c]   (di <= 3, out dim 64)
__global__ void k_in_linear(const float* __restrict__ feat,
                            const float* __restrict__ W,
                            const float* __restrict__ b,
                            float* __restrict__ h, int n, int di) {
  int t = blockIdx.x * blockDim.x + threadIdx.x;
  if (t >= n * 64) return;
  int v = t >> 6, c = t & 63;
  float acc = b[c];
  for (int k = 0; k < di; ++k) acc += feat[v * di + k] * W[k * 64 + c];
  h[(size_t)v * 64 + c] = acc;
}

// degree histograms for all 20 relations at once
__global__ void k_degree(const int* __restrict__ src, const int* __restrict__ dst,
                         float* __restrict__ degS, float* __restrict__ degD) {
  int t = blockIdx.x * blockDim.x + threadIdx.x;
  if (t >= 20 * NEDGE) return;
  int rel = t / NEDGE;
  atomAddF(&degS[(size_t)rel * 60000 + src[t]], 1.0f);
  atomAddF(&degD[(size_t)rel * 60000 + dst[t]], 1.0f);
}

// norm = (max(deg,1))^-1/2  in place
__global__ void k_norm(float* __restrict__ p, int n) {
  int t = blockIdx.x * blockDim.x + threadIdx.x;
  if (t < n) p[t] = rsqrtf(fmaxf(p[t], 1.0f));
}

// edge scatter: agg[dst] += h[src] * normS[src]   (DIN = 64 or 128)
template <int DIN>
__global__ void k_scatter(const float* __restrict__ h,
                          const float* __restrict__ normS,
                          const int* __restrict__ src,
                          const int* __restrict__ dst,
                          float* __restrict__ agg) {
  constexpr int C4 = DIN / 4;
  int t = blockIdx.x * blockDim.x + threadIdx.x;
  if (t >= NEDGE * C4) return;
  int e = t / C4;
  int c = (t - e * C4) * 4;
  int s = src[e], d = dst[e];
  float ns = normS[s];
  const float4 v = *(const float4*)(h + (size_t)s * DIN + c);
  float* ap = agg + (size_t)d * DIN + c;
  atomAddF(ap + 0, v.x * ns);
  atomAddF(ap + 1, v.y * ns);
  atomAddF(ap + 2, v.z * ns);
  atomAddF(ap + 3, v.w * ns);
}

// out[m][n] += (A @ W)[m][n] * normD[m] + bias[n]
// A: [Nd x DIN] fp32, W: [DIN x 128] fp32, out: [Nd x 128] fp32.
// One block = 32 rows x 128 cols; 8 waves, each wave owns one 16-col tile
// and TWO 16-row tiles, so each B fragment is loaded once and feeds two
// V_WMMA_F32_16X16X4_F32 issues (halves B L2 traffic vs 16-row blocks).
template <int DIN>
__global__ __launch_bounds__(256) void k_gemm(const float* __restrict__ A,
                                              const float* __restrict__ W,
                                              const float* __restrict__ bias,
                                              const float* __restrict__ normD,
                                              float* __restrict__ out) {
  constexpr int LDA = DIN + 4;          // pad: LDA % 64 == 4 -> conflict-free
  __shared__ float As[32 * LDA];

  const int m0 = blockIdx.x * 32;
  const int tid = threadIdx.x;

  // cooperative load of the 32 x DIN A-tile into LDS (float4)
  constexpr int C4 = DIN / 4;
  #pragma unroll
  for (int idx = tid; idx < 32 * C4; idx += 256) {
    int r = idx / C4;
    int c = (idx - r * C4) * 4;
    const float4 v = *(const float4*)(A + (size_t)(m0 + r) * DIN + c);
    float* s = &As[r * LDA + c];
    s[0] = v.x; s[1] = v.y; s[2] = v.z; s[3] = v.w;
  }
  __syncthreads();

  const int wave = tid >> 5;
  const int lane = tid & 31;
  const int half = lane >> 4;     // K-half selector per WMMA layout
  const int li   = lane & 15;     // A row within tile / B-D column
  const int n0   = wave * 16;

  v8f acc0 = {0.f, 0.f, 0.f, 0.f, 0.f, 0.f, 0.f, 0.f};
  v8f acc1 = {0.f, 0.f, 0.f, 0.f, 0.f, 0.f, 0.f, 0.f};

  #pragma unroll
  for (int k0 = 0; k0 < DIN; k0 += 4) {
    const int ka = k0 + 2 * half;
    // B 4x16 fp32 layout: N=li, K split across lane halves; loaded ONCE
    v2f b;
    b.x = W[(size_t)ka * HID + n0 + li];
    b.y = W[(size_t)(ka + 1) * HID + n0 + li];
    // A 16x4 fp32 layout for the two row-tiles
    v2f a0, a1;
    a0.x = As[li * LDA + ka];
    a0.y = As[li * LDA + ka + 1];
    a1.x = As[(li + 16) * LDA + ka];
    a1.y = As[(li + 16) * LDA + ka + 1];
    acc0 = __builtin_amdgcn_wmma_f32_16x16x4_f32(
        false, a0, false, b, (short)0, acc0, false, false);
    acc1 = __builtin_amdgcn_wmma_f32_16x16x4_f32(
        false, a1, false, b, (short)0, acc1, false, false);
  }

  // C/D layout: VGPR r holds (M = r + 8*half, N = li)
  #pragma unroll
  for (int r = 0; r < 8; ++r) {
    const int row0 = m0 + r + 8 * half;
    const int row1 = row0 + 16;
    const int col  = n0 + li;
    const float bb = bias[col];
    out[(size_t)row0 * HID + col] += acc0[r] * normD[row0] + bb;
    out[(size_t)row1 * HID + col] += acc1[r] * normD[row1] + bb;
  }
}

__global__ void k_relu(float* __restrict__ p, int n) {
  int t = blockIdx.x * blockDim.x + threadIdx.x;
  if (t < n) p[t] = fmaxf(p[t], 0.0f);
}

__global__ void k_out_init(float* __restrict__ o, const float* __restrict__ bf) {
  int t = threadIdx.x;
  if (t < 4) o[t] = bf[t];
}

// out[j] += sum_i Wf[j][i] * h[i]     (4 x 14.8M dot, grid-stride float4)
__global__ __launch_bounds__(256) void k_final(const float* __restrict__ Wf,
                                               const float* __restrict__ h,
                                               float* __restrict__ out) {
  const size_t n4 = (size_t)TOTAL_LEN / 4;
  float acc[4] = {0.f, 0.f, 0.f, 0.f};
  for (size_t i = (size_t)blockIdx.x * blockDim.x + threadIdx.x; i < n4;
       i += (size_t)gridDim.x * blockDim.x) {
    const float4 hv = *(const float4*)(h + i * 4);
    #pragma unroll
    for (int j = 0; j < 4; ++j) {
      const float4 wv = *(const float4*)(Wf + (size_t)j * TOTAL_LEN + i * 4);
      acc[j] += wv.x * hv.x + wv.y * hv.y + wv.z * hv.z + wv.w * hv.w;
    }
  }
  __shared__ float red[256];
  for (int j = 0; j < 4; ++j) {
    red[threadIdx.x] = acc[j];
    __syncthreads();
    for (int s = 128; s > 0; s >>= 1) {
      if (threadIdx.x < (unsigned)s) red[threadIdx.x] += red[threadIdx.x + s];
      __syncthreads();
    }
    if (threadIdx.x == 0) atomAddF(out + j, red[0]);
    __syncthreads();
  }
}

// ---------------- host orchestration ----------------
extern "C" void kernel_launch(void* const* d_in, const int* in_sizes, int n_in,
                              void* d_out, int out_size, void* d_ws, size_t ws_size,
                              hipStream_t stream) {
  (void)in_sizes; (void)n_in; (void)out_size; (void)ws_size;

  // inputs in setup_inputs() dict order
  const float* feat[NTYPES] = {(const float*)d_in[0], (const float*)d_in[3],
                               (const float*)d_in[6], (const float*)d_in[9],
                               (const float*)d_in[12]};
  const float* ltW[NTYPES]  = {(const float*)d_in[1], (const float*)d_in[4],
                               (const float*)d_in[7], (const float*)d_in[10],
                               (const float*)d_in[13]};
  const float* ltb[NTYPES]  = {(const float*)d_in[2], (const float*)d_in[5],
                               (const float*)d_in[8], (const float*)d_in[11],
                               (const float*)d_in[14]};
  const float* Wl3[3] = {(const float*)d_in[15], (const float*)d_in[17],
                         (const float*)d_in[19]};
  const float* bl3[3] = {(const float*)d_in[16], (const float*)d_in[18],
                         (const float*)d_in[20]};
  const float* Wf = (const float*)d_in[21];
  const float* bf = (const float*)d_in[22];
  const int* src = (const int*)d_in[23];
  const int* dst = (const int*)d_in[24];

  // node types: P, net, N, R, C
  static const int NNODE[NTYPES] = {20000, 60000, 20000, 8000, 8000};
  static const int NOFF[NTYPES]  = {0, 20000, 80000, 100000, 108000};
  static const int IND[NTYPES]   = {3, 1, 3, 1, 1};

  // relation table (src type, dst type), source order
  int relS[20], relD[20];
  for (int k = 0; k < 4; ++k) { relS[2*k] = 0; relD[2*k] = 1; relS[2*k+1] = 1; relD[2*k+1] = 0; }
  for (int k = 0; k < 4; ++k) { relS[8+2*k] = 2; relD[8+2*k] = 1; relS[8+2*k+1] = 1; relD[8+2*k+1] = 2; }
  relS[16] = 3; relD[16] = 1; relS[17] = 1; relD[17] = 3;
  relS[18] = 4; relD[18] = 1; relS[19] = 1; relD[19] = 4;

  // workspace carve-up (all float32)
  float* hA    = (float*)d_ws;                         // [116000 x 128] (layer-1 input uses width 64)
  float* hB    = hA + (size_t)TOTAL_NODES * HID;       // [116000 x 128]
  float* agg   = hB + (size_t)TOTAL_NODES * HID;       // [60000 x 128]
  float* normS = agg + (size_t)60000 * HID;            // [20 x 60000]
  float* normD = normS + (size_t)20 * 60000;           // [20 x 60000] (must follow normS)

  // 1) per-ntype input linear -> hA (width 64)
  for (int nt = 0; nt < NTYPES; ++nt) {
    int threads = NNODE[nt] * 64;
    k_in_linear<<<(threads + 255) / 256, 256, 0, stream>>>(
        feat[nt], ltW[nt], ltb[nt], hA + (size_t)NOFF[nt] * 64, NNODE[nt], IND[nt]);
  }

  // 2) degree norms (normS/normD contiguous -> one memset, one rsqrt pass)
  hipMemsetAsync(normS, 0, sizeof(float) * 2 * 20 * 60000, stream);
  k_degree<<<(20 * NEDGE + 255) / 256, 256, 0, stream>>>(src, dst, normS, normD);
  k_norm<<<(2 * 20 * 60000 + 255) / 256, 256, 0, stream>>>(normS, 2 * 20 * 60000);

  // 3) three hetero conv layers
  for (int l = 0; l < 3; ++l) {
    const int din = (l == 0) ? 64 : 128;
    const float* hin = (l == 1) ? hB : hA;   // L0: hA->hB, L1: hB->hA, L2: hA->hB
    float* hout = (l == 1) ? hA : hB;
    hipMemsetAsync(hout, 0, sizeof(float) * (size_t)TOTAL_NODES * HID, stream);

    for (int i = 0; i < 20; ++i) {
      const int s = relS[i], d = relD[i];
      const int nd = NNODE[d];
      hipMemsetAsync(agg, 0, sizeof(float) * (size_t)nd * din, stream);

      const float* hs   = hin + (size_t)NOFF[s] * din;
      const float* nS   = normS + (size_t)i * 60000;
      const float* nD   = normD + (size_t)i * 60000;
      const int*   se   = src + (size_t)i * NEDGE;
      const int*   de   = dst + (size_t)i * NEDGE;
      const float* Wrel = Wl3[l] + (size_t)i * din * HID;
      const float* brel = bl3[l] + (size_t)i * HID;
      float*       outp = hout + (size_t)NOFF[d] * HID;

      if (din == 64) {
        k_scatter<64><<<(NEDGE * 16 + 255) / 256, 256, 0, stream>>>(hs, nS, se, de, agg);
        k_gemm<64><<<nd / 32, 256, 0, stream>>>(agg, Wrel, brel, nD, outp);
      } else {
        k_scatter<128><<<(NEDGE * 32 + 255) / 256, 256, 0, stream>>>(hs, nS, se, de, agg);
        k_gemm<128><<<nd / 32, 256, 0, stream>>>(agg, Wrel, brel, nD, outp);
      }
    }
    k_relu<<<(TOTAL_LEN + 255) / 256, 256, 0, stream>>>(hout, TOTAL_LEN);
  }

  // 4) final 4 x TOTAL_LEN dot product (+ bias)
  k_out_init<<<1, 4, 0, stream>>>((float*)d_out, bf);
  k_final<<<2048, 256, 0, stream>>>(Wf, hB, (float*)d_out);
}